// AttGCNN_64046552318454
// MI455X (gfx1250) — compile-verified
//
#include <hip/hip_runtime.h>

#define N_NODES 20000
#define N_EDGES 320000
#define E_TOT   (N_EDGES + N_NODES)   // 340000 incl. self loops
#define IN_CH   128
#define HID     512                   // 4 heads * 128 ch
#define OUT_CH  4

typedef __attribute__((ext_vector_type(2))) float v2f;
typedef __attribute__((ext_vector_type(8))) float v8f;

// ---- helpers -------------------------------------------------------------
__device__ __forceinline__ unsigned enc_f(float f) {
  unsigned u = __float_as_uint(f);
  return (u & 0x80000000u) ? ~u : (u | 0x80000000u);   // monotone float->u32
}
__device__ __forceinline__ float dec_f(unsigned u) {
  unsigned b = (u & 0x80000000u) ? (u ^ 0x80000000u) : ~u;
  return __uint_as_float(b);
}
__device__ __forceinline__ void edge_sd(const int* __restrict__ ei, int e, int& s, int& d) {
  if (e < N_EDGES) { s = ei[e]; d = ei[N_EDGES + e]; }
  else             { s = e - N_EDGES; d = s; }          // appended self loop
}
__device__ __forceinline__ float lrelu(float x, float sl) { return x > 0.f ? x : sl * x; }

// ---- zero fill -----------------------------------------------------------
__global__ void k_zero(float* __restrict__ p, int n) {
  int i = blockIdx.x * blockDim.x + threadIdx.x;
  if (i < n) p[i] = 0.f;
}

// ---- layer1 GEMM: xl = x@Wl1, xr = x@Wr1 via V_WMMA_F32_16X16X4_F32 ------
// grid (1250, 8), block 128 (4 waves); wave w handles N-tile blockIdx.y*4+w
__global__ void k_gemm1(const float* __restrict__ x,
                        const float* __restrict__ Wl, const float* __restrict__ Wr,
                        float* __restrict__ xl, float* __restrict__ xr) {
  const int lane = threadIdx.x & 31;
  const int wave = threadIdx.x >> 5;
  const int m0   = blockIdx.x * 16;
  const int n0   = (blockIdx.y * 4 + wave) * 16;
  const int l16  = lane & 15;
  const int koff = (lane >> 4) << 1;             // 0 or 2
  const float* xrow = x + (size_t)(m0 + l16) * IN_CH;

  v8f accL = {0.f,0.f,0.f,0.f,0.f,0.f,0.f,0.f};
  v8f accR = {0.f,0.f,0.f,0.f,0.f,0.f,0.f,0.f};

  for (int k = 0; k < IN_CH; k += 4) {
    v2f a;  a.x  = xrow[k + koff];                 a.y  = xrow[k + koff + 1];
    v2f bl; bl.x = Wl[(k + koff) * HID + n0 + l16]; bl.y = Wl[(k + koff + 1) * HID + n0 + l16];
    v2f br; br.x = Wr[(k + koff) * HID + n0 + l16]; br.y = Wr[(k + koff + 1) * HID + n0 + l16];
    accL = __builtin_amdgcn_wmma_f32_16x16x4_f32(false, a, false, bl, (short)0, accL, false, false);
    accR = __builtin_amdgcn_wmma_f32_16x16x4_f32(false, a, false, br, (short)0, accR, false, false);
  }
  const int mrow = m0 + ((lane >> 4) << 3);
#pragma unroll
  for (int r = 0; r < 8; ++r) {
    xl[(size_t)(mrow + r) * HID + n0 + l16] = accL[r];
    xr[(size_t)(mrow + r) * HID + n0 + l16] = accR[r];
  }
}

// ---- layer1 edge scores + segment max (one wave per edge) ----------------
__global__ void k_score1(const float* __restrict__ xl, const float* __restrict__ xr,
                         const int* __restrict__ ei, const float* __restrict__ att,
                         float* __restrict__ score, unsigned* __restrict__ mmax) {
  int wid  = (blockIdx.x * blockDim.x + threadIdx.x) >> 5;
  int lane = threadIdx.x & 31;
  if (wid >= E_TOT) return;
  int s, d; edge_sd(ei, wid, s, d);
  const float* pl = xl + (size_t)s * HID;
  const float* pr = xr + (size_t)d * HID;
  const int c0 = lane * 16;                        // 16 contiguous ch per lane
  float partial = 0.f;
#pragma unroll
  for (int i = 0; i < 16; ++i) {
    int ch = c0 + i;                               // att[h][c] flat == att[ch]
    partial += att[ch] * lrelu(pl[ch] + pr[ch], 0.2f);
  }
  partial += __shfl_xor(partial, 1, 8);            // reduce 8 lanes = 1 head
  partial += __shfl_xor(partial, 2, 8);
  partial += __shfl_xor(partial, 4, 8);
  if ((lane & 7) == 0) {
    int h = lane >> 3;
    score[wid * 4 + h] = partial;
    atomicMax(&mmax[d * 4 + h], enc_f(partial));
  }
}

// ---- layer1 exp + segment sum (thread per edge-head) ---------------------
__global__ void k_expden1(float* __restrict__ score, const int* __restrict__ ei,
                          const unsigned* __restrict__ mmax, float* __restrict__ den) {
  int t = blockIdx.x * blockDim.x + threadIdx.x;
  if (t >= E_TOT * 4) return;
  int e = t >> 2, h = t & 3;
  int s, d; edge_sd(ei, e, s, d);
  float ex = __expf(score[t] - dec_f(mmax[d * 4 + h]));
  score[t] = ex;
  atomicAdd(&den[d * 4 + h], ex);
}

// ---- layer1 aggregation: acc[dst] += alpha * xl[src] (wave per edge) -----
__global__ void k_aggr1(const float* __restrict__ ex, const float* __restrict__ den,
                        const float* __restrict__ xl, const int* __restrict__ ei,
                        float* __restrict__ acc) {
  int wid  = (blockIdx.x * blockDim.x + threadIdx.x) >> 5;
  int lane = threadIdx.x & 31;
  if (wid >= E_TOT) return;
  int s, d; edge_sd(ei, wid, s, d);
  int h = lane >> 3;
  float alpha = ex[wid * 4 + h] / den[d * 4 + h];
  const float* pl = xl + (size_t)s * HID;
  float* pa = acc + (size_t)d * HID;
  const int c0 = lane * 16;
#pragma unroll
  for (int i = 0; i < 16; ++i)
    atomicAdd(&pa[c0 + i], alpha * pl[c0 + i]);
}

// ---- bias + LeakyReLU(0.01): h = act(acc + b1) ---------------------------
__global__ void k_biasact(const float* __restrict__ acc, const float* __restrict__ b,
                          float* __restrict__ h) {
  int t = blockIdx.x * blockDim.x + threadIdx.x;
  if (t >= N_NODES * HID) return;
  h[t] = lrelu(acc[t] + b[t & (HID - 1)], 0.01f);
}

// ---- layer2 GEMM (N=4, VALU): hl = h@Wl2, hr = h@Wr2 ---------------------
__global__ void k_gemm2(const float* __restrict__ h, const float* __restrict__ Wl,
                        const float* __restrict__ Wr,
                        float* __restrict__ hl, float* __restrict__ hr) {
  int t = blockIdx.x * blockDim.x + threadIdx.x;
  if (t >= N_NODES * OUT_CH) return;
  int n = t >> 2, j = t & 3;
  const float* hp = h + (size_t)n * HID;
  float al = 0.f, ar = 0.f;
  for (int k = 0; k < HID; ++k) {
    float v = hp[k];
    al += v * Wl[k * OUT_CH + j];
    ar += v * Wr[k * OUT_CH + j];
  }
  hl[t] = al; hr[t] = ar;
}

// ---- layer2 edge scores + max (thread per edge, 1 head, 4 ch) ------------
__global__ void k_score2(const float* __restrict__ hl, const float* __restrict__ hr,
                         const int* __restrict__ ei, const float* __restrict__ att,
                         float* __restrict__ score, unsigned* __restrict__ mmax) {
  int e = blockIdx.x * blockDim.x + threadIdx.x;
  if (e >= E_TOT) return;
  int s, d; edge_sd(ei, e, s, d);
  float acc = 0.f;
#pragma unroll
  for (int c = 0; c < 4; ++c)
    acc += att[c] * lrelu(hl[s * 4 + c] + hr[d * 4 + c], 0.2f);
  score[e] = acc;
  atomicMax(&mmax[d], enc_f(acc));
}

__global__ void k_expden2(float* __restrict__ score, const int* __restrict__ ei,
                          const unsigned* __restrict__ mmax, float* __restrict__ den) {
  int e = blockIdx.x * blockDim.x + threadIdx.x;
  if (e >= E_TOT) return;
  int s, d; edge_sd(ei, e, s, d);
  float ex = __expf(score[e] - dec_f(mmax[d]));
  score[e] = ex;
  atomicAdd(&den[d], ex);
}

__global__ void k_aggr2(const float* __restrict__ ex, const float* __restrict__ den,
                        const float* __restrict__ hl, const int* __restrict__ ei,
                        float* __restrict__ acc) {
  int e = blockIdx.x * blockDim.x + threadIdx.x;
  if (e >= E_TOT) return;
  int s, d; edge_sd(ei, e, s, d);
  float alpha = ex[e] / den[d];
#pragma unroll
  for (int c = 0; c < 4; ++c)
    atomicAdd(&acc[d * 4 + c], alpha * hl[s * 4 + c]);
}

// ---- bias + row softmax over 4 -------------------------------------------
__global__ void k_out(const float* __restrict__ acc, const float* __restrict__ b,
                      float* __restrict__ out) {
  int n = blockIdx.x * blockDim.x + threadIdx.x;
  if (n >= N_NODES) return;
  float v[4]; float mx = -3.0e38f;
#pragma unroll
  for (int c = 0; c < 4; ++c) { v[c] = acc[n * 4 + c] + b[c]; mx = fmaxf(mx, v[c]); }
  float sum = 0.f;
#pragma unroll
  for (int c = 0; c < 4; ++c) { v[c] = __expf(v[c] - mx); sum += v[c]; }
#pragma unroll
  for (int c = 0; c < 4; ++c) out[n * 4 + c] = v[c] / sum;
}

// ---- launcher ------------------------------------------------------------
extern "C" void kernel_launch(void* const* d_in, const int* in_sizes, int n_in,
                              void* d_out, int out_size, void* d_ws, size_t ws_size,
                              hipStream_t stream) {
  const float* x    = (const float*)d_in[0];
  const int*   ei   = (const int*)  d_in[1];   // [2, E] (src row, dst row)
  // d_in[2] = edge weights: unused by reference forward
  const float* Wl1  = (const float*)d_in[3];
  const float* Wr1  = (const float*)d_in[4];
  const float* att1 = (const float*)d_in[5];
  const float* b1   = (const float*)d_in[6];
  const float* Wl2  = (const float*)d_in[7];
  const float* Wr2  = (const float*)d_in[8];
  const float* att2 = (const float*)d_in[9];
  const float* b2   = (const float*)d_in[10];
  float* out = (float*)d_out;

  float* ws   = (float*)d_ws;
  float* XL   = ws;                                  // [N,512] xl -> later h
  float* XR   = XL  + (size_t)N_NODES * HID;         // [N,512] xr -> acc1
  float* SC1  = XR  + (size_t)N_NODES * HID;         // [E_TOT,4] score -> ex
  float* M1f  = SC1 + (size_t)E_TOT * 4;             // [N,4] enc max
  float* DEN1 = M1f + N_NODES * 4;                   // [N,4]
  float* HL2  = DEN1 + N_NODES * 4;                  // [N,4]
  float* HR2  = HL2 + N_NODES * 4;                   // [N,4]
  float* SC2  = HR2 + N_NODES * 4;                   // [E_TOT]
  float* M2f  = SC2 + E_TOT;                         // [N]
  float* DEN2 = M2f + N_NODES;                       // [N]
  float* ACC2 = DEN2 + N_NODES;                      // [N,4]
  unsigned* M1 = (unsigned*)M1f;
  unsigned* M2 = (unsigned*)M2f;

  const int smallN = 4*N_NODES + 4*N_NODES + 4*N_NODES + 4*N_NODES
                   + E_TOT + N_NODES + N_NODES + 4*N_NODES;   // M1..ACC2

  k_zero   <<<(smallN + 255) / 256, 256, 0, stream>>>(M1f, smallN);
  k_gemm1  <<<dim3(N_NODES / 16, HID / 64), 128, 0, stream>>>(x, Wl1, Wr1, XL, XR);
  k_score1 <<<(E_TOT + 7) / 8, 256, 0, stream>>>(XL, XR, ei, att1, SC1, M1);
  k_zero   <<<(N_NODES * HID + 255) / 256, 256, 0, stream>>>(XR, N_NODES * HID);
  k_expden1<<<(E_TOT * 4 + 255) / 256, 256, 0, stream>>>(SC1, ei, M1, DEN1);
  k_aggr1  <<<(E_TOT + 7) / 8, 256, 0, stream>>>(SC1, DEN1, XL, ei, XR);
  k_biasact<<<(N_NODES * HID + 255) / 256, 256, 0, stream>>>(XR, b1, XL);
  k_gemm2  <<<(N_NODES * OUT_CH + 255) / 256, 256, 0, stream>>>(XL, Wl2, Wr2, HL2, HR2);
  k_score2 <<<(E_TOT + 255) / 256, 256, 0, stream>>>(HL2, HR2, ei, att2, SC2, M2);
  k_expden2<<<(E_TOT + 255) / 256, 256, 0, stream>>>(SC2, ei, M2, DEN2);
  k_aggr2  <<<(E_TOT + 255) / 256, 256, 0, stream>>>(SC2, DEN2, HL2, ei, ACC2);
  k_out    <<<(N_NODES + 255) / 256, 256, 0, stream>>>(ACC2, b2, out);
}